// SimpleButEffectiveGNN_738734375048
// MI455X (gfx1250) — compile-verified
//
#include <hip/hip_runtime.h>
#include <hip/hip_bf16.h>

typedef float v2f __attribute__((ext_vector_type(2)));
typedef float v8f __attribute__((ext_vector_type(8)));

#define F_IN   128
#define H      32
#define NCLS   2
#define BN_EPS 1e-5f

// ---------------- atomics helpers ----------------
__device__ __forceinline__ void atomAddF(float* p, float v) {
    // hardware global_atomic_add_f32 (no CAS loop)
    unsafeAtomicAdd(p, v);
}

__device__ __forceinline__ void atomMaxF(float* p, float v) {
    // int-punned float max; requires init to -inf (0xFF800000)
    if (v >= 0.0f) atomicMax((int*)p, __float_as_int(v));
    else           atomicMin((unsigned int*)p, __float_as_uint(v));
}

// ---------------- init small buffers ----------------
__global__ void k_init(float* st1, float* st2, float* psum, float* pmax,
                       int* pcnt, int nG) {
    int i = blockIdx.x * blockDim.x + threadIdx.x;
    if (i < 128) { st1[i] = 0.0f; st2[i] = 0.0f; }
    if (i < nG * H) {
        psum[i] = 0.0f;
        pmax[i] = __int_as_float((int)0xFF800000);  // -inf
    }
    if (i < nG) pcnt[i] = 0;
}

// ---------------- agg1 = x (vectorized copy) ----------------
__global__ void k_copy4(const float4* __restrict__ src, float4* __restrict__ dst,
                        long n4) {
    long i = (long)blockIdx.x * blockDim.x + threadIdx.x;
    if (i < n4) dst[i] = src[i];
}

// ---------------- edge scatter, F=128 : one wave per edge ----------------
__global__ void k_edge_f128(const float* __restrict__ x, const int* __restrict__ src,
                            const int* __restrict__ dst, float* __restrict__ agg, int nE) {
    int e    = (int)((blockIdx.x * (long)blockDim.x + threadIdx.x) >> 5);
    int lane = threadIdx.x & 31;
    if (e >= nE) return;
    int s = src[e], d = dst[e];
    const float4 v = *(const float4*)(x + (long)s * F_IN + lane * 4);
    float* p = agg + (long)d * F_IN + lane * 4;
    atomAddF(p + 0, v.x); atomAddF(p + 1, v.y);
    atomAddF(p + 2, v.z); atomAddF(p + 3, v.w);
}

// ---------------- edge scatter, F=32 : one wave per edge ----------------
__global__ void k_edge_f32(const float* __restrict__ h, const int* __restrict__ src,
                           const int* __restrict__ dst, float* __restrict__ agg, int nE) {
    int e    = (int)((blockIdx.x * (long)blockDim.x + threadIdx.x) >> 5);
    int lane = threadIdx.x & 31;
    if (e >= nE) return;
    int s = src[e], d = dst[e];
    atomAddF(agg + (long)d * H + lane, h[(long)s * H + lane]);
}

// ---------------- GEMM [N,128] @ [128,32] + bias, fp32 WMMA ----------------
// one wave -> 16 rows x 32 cols; K loop in steps of 4 (V_WMMA_F32_16X16X4_F32)
__global__ void k_gemm_k128(const float* __restrict__ A, const float* __restrict__ W,
                            const float* __restrict__ bias, float* __restrict__ out,
                            int ntiles, int nrows) {
    int wv   = threadIdx.x >> 5;
    int lane = threadIdx.x & 31;
    int tile = blockIdx.x * 8 + wv;
    if (tile >= ntiles) return;                 // whole-wave exit: EXEC stays uniform
    int l15  = lane & 15;
    int half = lane >> 4;
    int rA   = tile * 16 + l15;
    if (rA >= nrows) rA = nrows - 1;            // clamp (stores are guarded)
    long rowA = (long)rA * F_IN;
    v8f acc0 = {}; v8f acc1 = {};
    for (int kb = 0; kb < F_IN; kb += 4) {
        int k = kb + half * 2;
        v2f a;  a.x  = A[rowA + k];              a.y  = A[rowA + k + 1];
        v2f b0; b0.x = W[k * H + l15];           b0.y = W[(k + 1) * H + l15];
        v2f b1; b1.x = W[k * H + 16 + l15];      b1.y = W[(k + 1) * H + 16 + l15];
        acc0 = __builtin_amdgcn_wmma_f32_16x16x4_f32(false, a, false, b0, (short)0, acc0, false, false);
        acc1 = __builtin_amdgcn_wmma_f32_16x16x4_f32(false, a, false, b1, (short)0, acc1, false, false);
    }
    long rbase = (long)tile * 16 + half * 8;
    float bi0 = bias[l15], bi1 = bias[16 + l15];
#pragma unroll
    for (int v = 0; v < 8; ++v) {
        long r = rbase + v;
        if (r < nrows) {
            out[r * H + l15]      = acc0[v] + bi0;
            out[r * H + 16 + l15] = acc1[v] + bi1;
        }
    }
}

// ---------------- GEMM [N,32] @ [32,32] + bias, fp32 WMMA ----------------
// optional fused BatchNorm+ReLU on the A operand, optional output ReLU,
// optional second destination (for seeding agg2 = h1).
template <bool BN_IN, bool RELU_OUT>
__global__ void k_gemm_k32(const float* __restrict__ A, const float* __restrict__ st,
                           const float* __restrict__ W, const float* __restrict__ bias,
                           float* __restrict__ out, float* __restrict__ out2,
                           int ntiles, int nrows) {
    int wv   = threadIdx.x >> 5;
    int lane = threadIdx.x & 31;
    int tile = blockIdx.x * 8 + wv;
    if (tile >= ntiles) return;
    int l15  = lane & 15;
    int half = lane >> 4;
    int rA   = tile * 16 + l15;
    if (rA >= nrows) rA = nrows - 1;
    long rowA = (long)rA * H;
    v8f acc0 = {}; v8f acc1 = {};
    for (int kb = 0; kb < H; kb += 4) {
        int k = kb + half * 2;
        float a0 = A[rowA + k], a1 = A[rowA + k + 1];
        if (BN_IN) {
            // st[64..95]=scale, st[96..127]=shift
            a0 = fmaxf(fmaf(a0, st[64 + k],     st[96 + k]),     0.0f);
            a1 = fmaxf(fmaf(a1, st[64 + k + 1], st[96 + k + 1]), 0.0f);
        }
        v2f a;  a.x  = a0;                        a.y  = a1;
        v2f b0; b0.x = W[k * H + l15];            b0.y = W[(k + 1) * H + l15];
        v2f b1; b1.x = W[k * H + 16 + l15];       b1.y = W[(k + 1) * H + 16 + l15];
        acc0 = __builtin_amdgcn_wmma_f32_16x16x4_f32(false, a, false, b0, (short)0, acc0, false, false);
        acc1 = __builtin_amdgcn_wmma_f32_16x16x4_f32(false, a, false, b1, (short)0, acc1, false, false);
    }
    long rbase = (long)tile * 16 + half * 8;
    float bi0 = bias[l15], bi1 = bias[16 + l15];
#pragma unroll
    for (int v = 0; v < 8; ++v) {
        long r = rbase + v;
        if (r < nrows) {
            float o0 = acc0[v] + bi0;
            float o1 = acc1[v] + bi1;
            if (RELU_OUT) { o0 = fmaxf(o0, 0.0f); o1 = fmaxf(o1, 0.0f); }
            out[r * H + l15]      = o0;
            out[r * H + 16 + l15] = o1;
            if (out2) {
                out2[r * H + l15]      = o0;
                out2[r * H + 16 + l15] = o1;
            }
        }
    }
}

// ---------------- per-channel sum / sumsq for BatchNorm ----------------
__global__ void k_bn_stats(const float* __restrict__ h, float* __restrict__ st,
                           int nrows) {
    __shared__ float s1[256];
    __shared__ float s2[256];
    int ch = threadIdx.x & 31;
    int rg = threadIdx.x >> 5;                  // 8 row-lanes per block
    float a = 0.0f, b = 0.0f;
    for (long r = (long)blockIdx.x * 8 + rg; r < nrows; r += (long)gridDim.x * 8) {
        float v = h[r * H + ch];
        a += v; b += v * v;
    }
    s1[threadIdx.x] = a; s2[threadIdx.x] = b;
    __syncthreads();
    if (threadIdx.x < 32) {
        float ta = 0.0f, tb = 0.0f;
#pragma unroll
        for (int j = 0; j < 8; ++j) { ta += s1[threadIdx.x + 32 * j]; tb += s2[threadIdx.x + 32 * j]; }
        atomAddF(&st[threadIdx.x], ta);
        atomAddF(&st[32 + threadIdx.x], tb);
    }
}

// ---------------- fold stats -> scale/shift ----------------
__global__ void k_bn_finalize(float* st, const float* __restrict__ gamma,
                              const float* __restrict__ beta, float inv_n) {
    int c = threadIdx.x;
    if (c < H) {
        float mu  = st[c] * inv_n;
        float var = st[32 + c] * inv_n - mu * mu;
        float sc  = gamma[c] * rsqrtf(var + BN_EPS);
        st[64 + c] = sc;
        st[96 + c] = beta[c] - mu * sc;
    }
}

// ---------------- per-graph mean/max pooling ----------------
__global__ void k_pool(const float* __restrict__ h2, const int* __restrict__ batch,
                       float* __restrict__ psum, float* __restrict__ pmax,
                       int* __restrict__ pcnt, int nrows) {
    long idx = (long)blockIdx.x * blockDim.x + threadIdx.x;
    if (idx >= (long)nrows * H) return;
    int node = (int)(idx >> 5);
    int ch   = (int)(idx & 31);
    int g    = batch[node];
    float v  = h2[idx];
    atomAddF(&psum[(long)g * H + ch], v);
    atomMaxF(&pmax[(long)g * H + ch], v);
    if (ch == 0) atomicAdd(&pcnt[g], 1);
}

// ---------------- classifier: [G,64] -> relu -> [G,2] ----------------
__global__ void k_classifier(const float* __restrict__ psum, const float* __restrict__ pmax,
                             const int* __restrict__ pcnt,
                             const float* __restrict__ Wc1, const float* __restrict__ bc1,
                             const float* __restrict__ Wc2, const float* __restrict__ bc2,
                             float* __restrict__ out, int nG) {
    int g = blockIdx.x * blockDim.x + threadIdx.x;
    if (g >= nG) return;
    int   cnt = pcnt[g];
    float inv = 1.0f / fmaxf((float)cnt, 1.0f);
    float z[2 * H];
#pragma unroll
    for (int c = 0; c < H; ++c) {
        z[c]     = psum[(long)g * H + c] * inv;
        float m  = pmax[(long)g * H + c];
        z[H + c] = (cnt > 0) ? m : 0.0f;
    }
    float hid[H];
    for (int j = 0; j < H; ++j) {
        float acc = bc1[j];
        for (int k = 0; k < 2 * H; ++k) acc = fmaf(z[k], Wc1[k * H + j], acc);
        hid[j] = fmaxf(acc, 0.0f);
    }
    for (int c = 0; c < NCLS; ++c) {
        float acc = bc2[c];
        for (int j = 0; j < H; ++j) acc = fmaf(hid[j], Wc2[j * NCLS + c], acc);
        out[(long)g * NCLS + c] = acc;
    }
}

// =======================================================================
extern "C" void kernel_launch(void* const* d_in, const int* in_sizes, int n_in,
                              void* d_out, int out_size, void* d_ws, size_t ws_size,
                              hipStream_t stream) {
    const float* x   = (const float*)d_in[0];
    const int*   ei  = (const int*)d_in[1];
    const int*   bat = (const int*)d_in[2];
    const float* W1  = (const float*)d_in[3];
    const float* b1  = (const float*)d_in[4];
    const float* g1  = (const float*)d_in[5];
    const float* be1 = (const float*)d_in[6];
    const float* W2  = (const float*)d_in[7];
    const float* b2  = (const float*)d_in[8];
    const float* W3  = (const float*)d_in[9];
    const float* b3  = (const float*)d_in[10];
    const float* g2  = (const float*)d_in[11];
    const float* be2 = (const float*)d_in[12];
    const float* W4  = (const float*)d_in[13];
    const float* b4  = (const float*)d_in[14];
    const float* Wc1 = (const float*)d_in[15];
    const float* bc1 = (const float*)d_in[16];
    const float* Wc2 = (const float*)d_in[17];
    const float* bc2 = (const float*)d_in[18];

    const int nN = in_sizes[0] / F_IN;     // 100000
    const int nE = in_sizes[1] / 2;        // 1600000
    const int nG = out_size / NCLS;        // 1000
    const int* src = ei;
    const int* dst = ei + nE;

    // workspace layout (floats)
    float* ws   = (float*)d_ws;
    size_t o    = 0;
    float* agg1 = ws + o; o += (size_t)nN * F_IN;  // [N,128]
    float* pre  = ws + o; o += (size_t)nN * H;     // h1pre / h2pre (reused)
    float* h1   = ws + o; o += (size_t)nN * H;
    float* agg2 = ws + o; o += (size_t)nN * H;
    float* h2   = ws + o; o += (size_t)nN * H;
    float* st1  = ws + o; o += 128;                // sum|sumsq|scale|shift
    float* st2  = ws + o; o += 128;
    float* psum = ws + o; o += (size_t)nG * H;
    float* pmax = ws + o; o += (size_t)nG * H;
    int*   pcnt = (int*)(ws + o); o += nG;

    const int ntiles  = (nN + 15) / 16;
    const int gblocks = (ntiles + 7) / 8;          // 8 waves / block
    const int eblocks = (nE + 7) / 8;              // 8 edges (waves) / block
    const float inv_n = 1.0f / (float)nN;

    // 0) zero/seed small buffers
    k_init<<<(nG * H + 255) / 256, 256, 0, stream>>>(st1, st2, psum, pmax, pcnt, nG);

    // ---- layer 1 ----
    long n4 = (long)nN * F_IN / 4;
    k_copy4<<<(int)((n4 + 255) / 256), 256, 0, stream>>>((const float4*)x, (float4*)agg1, n4);
    k_edge_f128<<<eblocks, 256, 0, stream>>>(x, src, dst, agg1, nE);
    k_gemm_k128<<<gblocks, 256, 0, stream>>>(agg1, W1, b1, pre, ntiles, nN);
    k_bn_stats<<<512, 256, 0, stream>>>(pre, st1, nN);
    k_bn_finalize<<<1, 32, 0, stream>>>(st1, g1, be1, inv_n);
    // BN+ReLU fused into A-load; outer ReLU fused into store; also seeds agg2=h1
    k_gemm_k32<true, true><<<gblocks, 256, 0, stream>>>(pre, st1, W2, b2, h1, agg2, ntiles, nN);

    // ---- layer 2 ----
    k_edge_f32<<<eblocks, 256, 0, stream>>>(h1, src, dst, agg2, nE);
    k_gemm_k32<false, false><<<gblocks, 256, 0, stream>>>(agg2, nullptr, W3, b3, pre, nullptr, ntiles, nN);
    k_bn_stats<<<512, 256, 0, stream>>>(pre, st2, nN);
    k_bn_finalize<<<1, 32, 0, stream>>>(st2, g2, be2, inv_n);
    k_gemm_k32<true, false><<<gblocks, 256, 0, stream>>>(pre, st2, W4, b4, h2, nullptr, ntiles, nN);

    // ---- pooling + classifier ----
    k_pool<<<(int)(((long)nN * H + 255) / 256), 256, 0, stream>>>(h2, bat, psum, pmax, pcnt, nN);
    k_classifier<<<(nG + 255) / 256, 256, 0, stream>>>(psum, pmax, pcnt,
                                                       Wc1, bc1, Wc2, bc2,
                                                       (float*)d_out, nG);
}